// CausalConv1d_10479720202642
// MI455X (gfx1250) — compile-verified
//
#include <hip/hip_runtime.h>
#include <stdint.h>

typedef float f32x4 __attribute__((ext_vector_type(4)));
typedef int   v4i   __attribute__((vector_size(16)));

#define TPB   256          // threads per block (8 waves, wave32)
#define VEC   4            // channels per thread (float4)
#define WD    (TPB * VEC)  // 1024 channels per block strip
#define CT    64           // time rows per block chunk
#define KTAPS 4
#define PRE   12           // async prefetch depth (rows in flight per wave)
#define NST   16           // LDS ring slots (power of two, > PRE + window)

// Fixed problem shape (from the reference): B=4, T=4096, D=2048
#define TDIM  4096
#define DDIM  2048
#define NCHUNK_T (TDIM / CT)   // 64  (power of two -> shift/mask decode)
#define NSTRIP_D (DDIM / WD)   // 2

#if defined(__HIP_DEVICE_COMPILE__) && __has_builtin(__builtin_amdgcn_global_load_async_to_lds_b128)
#define HAVE_ASYNC 1
#else
#define HAVE_ASYNC 0
#endif

#if defined(__HIP_DEVICE_COMPILE__) && __has_builtin(__builtin_amdgcn_s_wait_asynccnt)
#define WAIT_ASYNC(n) __builtin_amdgcn_s_wait_asynccnt(n)
#elif HAVE_ASYNC
#define WAIT_ASYNC(n) asm volatile("s_wait_asynccnt %0" ::"i"(n) : "memory")
#else
#define WAIT_ASYNC(n)
#endif

#if HAVE_ASYNC
#define ASYNC_LD_B128(gsrc, ldst)                                     \
    __builtin_amdgcn_global_load_async_to_lds_b128(                   \
        (__attribute__((address_space(1))) v4i*)(gsrc),               \
        (__attribute__((address_space(3))) v4i*)(ldst), 0, 0)
#endif

// y[b,t,d] = bias[d] + sum_{k=0..3} w[d,k] * x[b, t-3+k, d]   (zero pad t<0)
__global__ __launch_bounds__(TPB) void causal_dwconv4_kernel(
    const float* __restrict__ x,     // (B,T,D)
    const float* __restrict__ w,     // (D,1,4)
    const float* __restrict__ bias,  // (D,)
    float* __restrict__ y)           // (B,T,D)
{
#if HAVE_ASYNC
    __shared__ float lds[NST * WD];  // 16 * 4 KiB = 64 KiB ring of x rows
#endif
    int bid = blockIdx.x;
    const int chunk = bid & (NCHUNK_T - 1);  bid >>= 6;   // log2(NCHUNK_T)=6
    const int strip = bid & (NSTRIP_D - 1);  bid >>= 1;   // log2(NSTRIP_D)=1
    const int b     = bid;
    const int tid   = threadIdx.x;
    const int d0    = strip * WD + tid * VEC;   // this thread's 4 channels
    const int t0    = chunk * CT;
    const int tstart = t0 - (KTAPS - 1);
    const int tlast  = t0 + CT - 1;

    // Load this thread's taps (w is (D,1,4): taps of channel d are contiguous)
    // and transpose so wk[k] is a 4-channel vector for tap k.
    f32x4 w0, w1, w2, w3;
    {
        const f32x4* wp = (const f32x4*)(w + (size_t)d0 * KTAPS);
        f32x4 c0 = wp[0], c1 = wp[1], c2 = wp[2], c3 = wp[3];
        w0 = (f32x4){c0.x, c1.x, c2.x, c3.x};
        w1 = (f32x4){c0.y, c1.y, c2.y, c3.y};
        w2 = (f32x4){c0.z, c1.z, c2.z, c3.z};
        w3 = (f32x4){c0.w, c1.w, c2.w, c3.w};
    }
    const f32x4 bv = *(const f32x4*)(bias + d0);

    const float* xb = x + (size_t)b * TDIM * DDIM;
    float*       yb = y + (size_t)b * TDIM * DDIM;

    // Sliding window registers: rows r-3, r-2, r-1
    f32x4 x0 = (f32x4)0.0f, x1 = (f32x4)0.0f, x2 = (f32x4)0.0f;

#if HAVE_ASYNC
    // Prologue: fill the async pipeline with PRE rows. Negative rows are
    // address-clamped (never consumed) so ASYNCcnt accounting stays uniform.
#pragma unroll
    for (int q = tstart; q < tstart + PRE; ++q) {
        int qc = q < 0 ? 0 : (q > tlast ? tlast : q);
        ASYNC_LD_B128(xb + (size_t)qc * DDIM + d0,
                      &lds[((q - tstart) & (NST - 1)) * WD + tid * VEC]);
    }
    // Window fill: rows t0-3 .. t0-1 (zero for t<0), peeled out of hot loop.
#pragma unroll
    for (int r = tstart; r < t0; ++r) {
        WAIT_ASYNC(PRE - 1);
        asm volatile("" ::: "memory");
        f32x4 xn;
        if (r >= 0) {                        // wave-uniform branch
            xn = *(const f32x4*)(&lds[((r - tstart) & (NST - 1)) * WD + tid * VEC]);
        } else {
            xn = (f32x4)0.0f;                // causal zero padding
        }
        int q = r + PRE;                     // q >= t0+PRE-3 >= 9 > 0
        ASYNC_LD_B128(xb + (size_t)q * DDIM + d0,
                      &lds[((q - tstart) & (NST - 1)) * WD + tid * VEC]);
        x0 = x1; x1 = x2; x2 = xn;
    }
    // Hot loop: branch-free steady state, one row in / one row out.
#pragma unroll 4
    for (int r = t0; r <= tlast; ++r) {
        WAIT_ASYNC(PRE - 1);                 // row r's DMA has retired
        asm volatile("" ::: "memory");       // keep LDS read below the wait
        f32x4 xn = *(const f32x4*)(&lds[((r - tstart) & (NST - 1)) * WD + tid * VEC]);
        {   // refill: row r+PRE (clamped at the chunk tail; slot r+PRE-NST
            // held row r-4 which was consumed 4 iterations ago)
            int q  = r + PRE;
            int qc = q > tlast ? tlast : q;
            ASYNC_LD_B128(xb + (size_t)qc * DDIM + d0,
                          &lds[((q - tstart) & (NST - 1)) * WD + tid * VEC]);
        }
        f32x4 acc = bv;
        acc += w0 * x0;   // tap 0 -> x[r-3]
        acc += w1 * x1;   // tap 1 -> x[r-2]
        acc += w2 * x2;   // tap 2 -> x[r-1]
        acc += w3 * xn;   // tap 3 -> x[r]
        __builtin_nontemporal_store(acc, (f32x4*)(yb + (size_t)r * DDIM + d0));
        x0 = x1; x1 = x2; x2 = xn;
    }
#else
    // Fallback path: direct b128 loads + explicit prefetch.
    for (int r = tstart; r <= tlast; ++r) {
        f32x4 xn;
        if (r >= 0) {
            xn = *(const f32x4*)(xb + (size_t)r * DDIM + d0);
        } else {
            xn = (f32x4)0.0f;
        }
        int q  = r + PRE;
        int qc = q > tlast ? tlast : q;
        __builtin_prefetch(xb + (size_t)qc * DDIM + d0, 0, 0);  // global_prefetch_b8
        if (r >= t0) {
            f32x4 acc = bv;
            acc += w0 * x0;
            acc += w1 * x1;
            acc += w2 * x2;
            acc += w3 * xn;
            __builtin_nontemporal_store(acc, (f32x4*)(yb + (size_t)r * DDIM + d0));
        }
        x0 = x1; x1 = x2; x2 = xn;
    }
#endif
}

extern "C" void kernel_launch(void* const* d_in, const int* in_sizes, int n_in,
                              void* d_out, int out_size, void* d_ws, size_t ws_size,
                              hipStream_t stream) {
    const float* x = (const float*)d_in[0];
    const float* w = (const float*)d_in[1];
    const float* b = (const float*)d_in[2];
    float*       y = (float*)d_out;

    const int D  = DDIM;                    // 2048 (== in_sizes[2])
    const int BT = in_sizes[0] / D;         // B*T
    const int B  = BT / TDIM;               // 4

    dim3 grid(B * NSTRIP_D * NCHUNK_T);     // 4 * 2 * 64 = 512 blocks
    causal_dwconv4_kernel<<<grid, TPB, 0, stream>>>(x, w, b, y);
}